// _AnchorTargetLayer_89412629168452
// MI455X (gfx1250) — compile-verified
//
#include <hip/hip_runtime.h>
#include <hip/hip_bf16.h>
#include <math.h>

// ---------------------------------------------------------------------------
// RPN AnchorTargetLayer for MI455X (gfx1250, wave32).
// No matmul content -> no WMMA. CDNA5 paths used: async global->LDS staging of
// the reused gt-box / base-anchor tables (global_load_async_to_lds_b32 +
// s_wait_asynccnt), wave32-safe block reductions, LDS histogram atomics.
// This round: replace runtime-divisor integer divides in the hot anchor
// decode with shift/mask (pow2 fast path), which matters most in k1b_gtmax
// where the decode is per-IoU-pair.
// ---------------------------------------------------------------------------

#define NBINS     2048
#define MAXG      64
#define MAXA      16
#define CAND_CAP  4096
#define NUM_FG_K  64          // 0.25 * 256
#define RPN_BATCH 256
#define POS_OVL   0.5f
#define NEG_OVL   0.3f
#define NEG_INF_F (-1e30f)
#define FSTRIDE   16.0f
#define GPARTS    16

#ifndef __has_builtin
#define __has_builtin(x) 0
#endif
#if __has_builtin(__builtin_amdgcn_global_load_async_to_lds_b32) && \
    __has_builtin(__builtin_amdgcn_s_wait_asynccnt)
#define HAVE_ASYNC_LDS 1
#else
#define HAVE_ASYNC_LDS 0
#endif

#define AS1 __attribute__((address_space(1)))
#define AS3 __attribute__((address_space(3)))

#define DEV __device__ __forceinline__

// Stage n floats from global into LDS using the gfx1250 async-to-LDS path.
DEV void stage_lds(float* lds, const float* gp, int n) {
  for (int i = (int)threadIdx.x; i < n; i += (int)blockDim.x) {
#if HAVE_ASYNC_LDS
    __builtin_amdgcn_global_load_async_to_lds_b32(
        (AS1 int*)(gp + i), (AS3 int*)(lds + i), 0, 0);
#else
    lds[i] = gp[i];
#endif
  }
}
DEV void wait_stage() {
#if HAVE_ASYNC_LDS
  __builtin_amdgcn_s_wait_asynccnt(0);
#endif
}

// Monotonic float<->uint mapping so we can use integer atomicMax for fp max.
DEV unsigned enc_f32(float f) {
  unsigned u = __float_as_uint(f);
  return (u & 0x80000000u) ? ~u : (u | 0x80000000u);
}
DEV float dec_f32(unsigned e) {
  unsigned b = (e & 0x80000000u) ? (e & 0x7FFFFFFFu) : ~e;
  return __uint_as_float(b);
}

// IoU for one (anchor, gt) pair. Uses explicit _rn intrinsics so that the
// value is bit-identical across the three kernels that recompute it (needed
// for the exact `iou == gt_max` "best anchor" test).
DEV float iou_pair(float ax1, float ay1, float ax2, float ay2, float aarea,
                   float g0, float g1, float g2, float g3, bool inside) {
  float gw = __fadd_rn(__fsub_rn(g2, g0), 1.0f);
  float gh = __fadd_rn(__fsub_rn(g3, g1), 1.0f);
  float ix1 = fmaxf(ax1, g0), iy1 = fmaxf(ay1, g1);
  float ix2 = fminf(ax2, g2), iy2 = fminf(ay2, g3);
  float iw = fmaxf(__fadd_rn(__fsub_rn(ix2, ix1), 1.0f), 0.0f);
  float ih = fmaxf(__fadd_rn(__fsub_rn(iy2, iy1), 1.0f), 0.0f);
  float inter = __fmul_rn(iw, ih);
  float garea = __fmul_rn(gw, gh);
  float denom = __fsub_rn(__fadd_rn(aarea, garea), inter);
  float v = inter / denom;
  if (gw == 1.0f && gh == 1.0f) v = 0.0f;   // zero-padded gt
  if (!inside) v = -1.0f;                    // outside anchors excluded
  return v;
}

// Anchor decode: index m = ((a*H)+y)*W + x (x fastest). hwsh/wsh are log2 of
// HW/W when they are powers of two (else -1 -> divide fallback). The branch
// is on an SGPR-uniform value, so it is a cheap scalar branch.
DEV void decode_m(int m, int HW, int W, int hwsh, int wsh,
                  int& a, int& y, int& x) {
  int r;
  if (hwsh >= 0) { a = m >> hwsh; r = m & (HW - 1); }
  else           { a = m / HW;    r = m - a * HW;   }
  if (wsh >= 0)  { y = r >> wsh;  x = r & (W - 1);  }
  else           { y = r / W;     x = r - y * W;    }
}

DEV void anchor_box(const float* sanc, int a, int x, int y,
                    float& ax1, float& ay1, float& ax2, float& ay2,
                    float& aw, float& ah, float& aarea) {
  float fx = (float)x * FSTRIDE, fy = (float)y * FSTRIDE;
  ax1 = __fadd_rn(sanc[a * 4 + 0], fx);
  ay1 = __fadd_rn(sanc[a * 4 + 1], fy);
  ax2 = __fadd_rn(sanc[a * 4 + 2], fx);
  ay2 = __fadd_rn(sanc[a * 4 + 3], fy);
  aw = __fadd_rn(__fsub_rn(ax2, ax1), 1.0f);
  ah = __fadd_rn(__fsub_rn(ay2, ay1), 1.0f);
  aarea = __fmul_rn(aw, ah);
}

// ---------------------------------------------------------------------------
// K0: zero the accumulators (gt_max encodings + per-image histograms).
// ---------------------------------------------------------------------------
__global__ void k0_init(unsigned* gtmax, unsigned* hist, int ngt, int nhist) {
  int i = blockIdx.x * blockDim.x + threadIdx.x;
  if (i < ngt) gtmax[i] = 0u;          // enc of absolute-minimum key
  if (i < nhist) hist[i] = 0u;
}

// ---------------------------------------------------------------------------
// K1: per anchor -> max IoU over gts + argmax (first occurrence).
// grid = (ceil(NA/256), B)
// ---------------------------------------------------------------------------
__global__ void k1_maxarg(const float* __restrict__ gt,
                          const float* __restrict__ anchors,
                          const float* __restrict__ im_info,
                          float* __restrict__ maxov, int* __restrict__ argm,
                          int G, int A, int H, int W, int hwsh, int wsh) {
  __shared__ float sgt[MAXG * 5];
  __shared__ float sanc[MAXA * 4];
  int b = blockIdx.y;
  stage_lds(sgt, gt + (size_t)b * G * 5, G * 5);
  stage_lds(sanc, anchors, A * 4);
  wait_stage();
  __syncthreads();

  int HW = H * W, NA = A * HW;
  int m = blockIdx.x * blockDim.x + threadIdx.x;
  if (m >= NA) return;
  int a, y, x; decode_m(m, HW, W, hwsh, wsh, a, y, x);
  float ax1, ay1, ax2, ay2, aw, ah, aarea;
  anchor_box(sanc, a, x, y, ax1, ay1, ax2, ay2, aw, ah, aarea);
  float im_h = im_info[0], im_w = im_info[1];
  bool inside = (ax1 >= 0.0f) && (ay1 >= 0.0f) && (ax2 < im_w) && (ay2 < im_h);

  float best = -2.0f; int bi = 0;
  for (int g = 0; g < G; ++g) {
    float v = iou_pair(ax1, ay1, ax2, ay2, aarea,
                       sgt[g * 5 + 0], sgt[g * 5 + 1], sgt[g * 5 + 2], sgt[g * 5 + 3],
                       inside);
    if (v > best) { best = v; bi = g; }
  }
  size_t bm = (size_t)b * NA + m;
  maxov[bm] = best;
  argm[bm] = bi;
}

// ---------------------------------------------------------------------------
// K1b: per (b, gt) -> column max over all anchors (gt_max), via block reduce
// + one global atomicMax per block. grid = B*G*GPARTS blocks.
// ---------------------------------------------------------------------------
__global__ void k1b_gtmax(const float* __restrict__ gt,
                          const float* __restrict__ anchors,
                          const float* __restrict__ im_info,
                          unsigned* __restrict__ gtmax,
                          int G, int A, int H, int W, int hwsh, int wsh) {
  __shared__ float sanc[MAXA * 4];
  __shared__ float red[256];
  int id = blockIdx.x;
  int p = id % GPARTS; id /= GPARTS;
  int g = id % G;
  int b = id / G;
  for (int i = threadIdx.x; i < A * 4; i += blockDim.x) sanc[i] = anchors[i];
  __syncthreads();

  float g0 = gt[((size_t)b * G + g) * 5 + 0];
  float g1 = gt[((size_t)b * G + g) * 5 + 1];
  float g2 = gt[((size_t)b * G + g) * 5 + 2];
  float g3 = gt[((size_t)b * G + g) * 5 + 3];
  float im_h = im_info[0], im_w = im_info[1];

  int HW = H * W, NA = A * HW;
  float lm = -2.0f;
  for (int m = p * blockDim.x + threadIdx.x; m < NA; m += GPARTS * blockDim.x) {
    int a, y, x; decode_m(m, HW, W, hwsh, wsh, a, y, x);
    float ax1, ay1, ax2, ay2, aw, ah, aarea;
    anchor_box(sanc, a, x, y, ax1, ay1, ax2, ay2, aw, ah, aarea);
    bool inside = (ax1 >= 0.0f) && (ay1 >= 0.0f) && (ax2 < im_w) && (ay2 < im_h);
    float v = iou_pair(ax1, ay1, ax2, ay2, aarea, g0, g1, g2, g3, inside);
    lm = fmaxf(lm, v);
  }
  red[threadIdx.x] = lm;
  __syncthreads();
  for (int s = blockDim.x / 2; s > 0; s >>= 1) {
    if ((int)threadIdx.x < s) red[threadIdx.x] = fmaxf(red[threadIdx.x], red[threadIdx.x + s]);
    __syncthreads();
  }
  if (threadIdx.x == 0) atomicMax(&gtmax[b * G + g], enc_f32(red[0]));
}

// ---------------------------------------------------------------------------
// K2: labels (pre-subsample) + per-image fg/bg score histograms (LDS->global).
// grid = (ceil(NA/256), B)
// ---------------------------------------------------------------------------
__global__ void k2_labels(const float* __restrict__ gt,
                          const float* __restrict__ anchors,
                          const float* __restrict__ im_info,
                          const float* __restrict__ maxov,
                          const unsigned* __restrict__ gtmaxe,
                          const float* __restrict__ rs,
                          int* __restrict__ labels,
                          unsigned* __restrict__ hist,
                          int G, int A, int H, int W, int hwsh, int wsh) {
  __shared__ float sgt[MAXG * 5];
  __shared__ float sanc[MAXA * 4];
  __shared__ float sgm[MAXG];
  __shared__ unsigned hfg[NBINS];
  __shared__ unsigned hbg[NBINS];
  int b = blockIdx.y;
  stage_lds(sgt, gt + (size_t)b * G * 5, G * 5);
  stage_lds(sanc, anchors, A * 4);
  for (int i = threadIdx.x; i < NBINS; i += blockDim.x) { hfg[i] = 0u; hbg[i] = 0u; }
  for (int i = threadIdx.x; i < G; i += blockDim.x) {
    float v = dec_f32(gtmaxe[b * G + i]);
    sgm[i] = (v == 0.0f) ? 1e-5f : v;
  }
  wait_stage();
  __syncthreads();

  int HW = H * W, NA = A * HW;
  int m = blockIdx.x * blockDim.x + threadIdx.x;
  if (m < NA) {
    int a, y, x; decode_m(m, HW, W, hwsh, wsh, a, y, x);
    float ax1, ay1, ax2, ay2, aw, ah, aarea;
    anchor_box(sanc, a, x, y, ax1, ay1, ax2, ay2, aw, ah, aarea);
    float im_h = im_info[0], im_w = im_info[1];
    bool inside = (ax1 >= 0.0f) && (ay1 >= 0.0f) && (ax2 < im_w) && (ay2 < im_h);
    float mo = maxov[(size_t)b * NA + m];
    int cnt = 0;
    for (int g = 0; g < G; ++g) {
      float v = iou_pair(ax1, ay1, ax2, ay2, aarea,
                         sgt[g * 5 + 0], sgt[g * 5 + 1], sgt[g * 5 + 2], sgt[g * 5 + 3],
                         inside);
      cnt += (v == sgm[g]) ? 1 : 0;
    }
    // reference order: best->1, pos->1, neg->0, outside->-1
    int l = -1;
    if (cnt > 0) l = 1;
    if (mo >= POS_OVL) l = 1;
    if (mo < NEG_OVL) l = 0;
    if (!inside) l = -1;
    labels[(size_t)b * NA + m] = l;
    if (l == 1 || l == 0) {
      float s = rs[(size_t)b * NA + (size_t)(y * W + x) * A + a];
      int bin = (int)(s * (float)NBINS);
      if (bin > NBINS - 1) bin = NBINS - 1;
      if (bin < 0) bin = 0;
      if (l == 1) atomicAdd(&hfg[bin], 1u);
      else        atomicAdd(&hbg[bin], 1u);
    }
  }
  __syncthreads();
  unsigned* gH = hist + (size_t)b * 2 * NBINS;
  for (int i = threadIdx.x; i < NBINS; i += blockDim.x) {
    if (hfg[i]) atomicAdd(&gH[i], hfg[i]);
    if (hbg[i]) atomicAdd(&gH[NBINS + i], hbg[i]);
  }
}

// ---------------------------------------------------------------------------
// K3: per (image, class) exact k-th-largest score threshold via histogram
// bucket + exact in-bucket selection. grid = (B, 2), 1024 threads.
// Also computes uni_w = 1/num_examples (image 0) analytically.
// ---------------------------------------------------------------------------
__global__ void k3_thresh(const unsigned* __restrict__ hist,
                          const float* __restrict__ rs,
                          const int* __restrict__ labels,
                          float* __restrict__ thr,
                          float* __restrict__ uniw,
                          int A, int H, int W, int hwsh, int wsh) {
  int b = blockIdx.x, cls = blockIdx.y;   // cls 0 = fg, 1 = bg
  __shared__ int s_bucket, s_kprime, s_done;
  __shared__ unsigned s_cnt;
  __shared__ float cand[CAND_CAP];
  if (threadIdx.x == 0) {
    const unsigned* hf = hist + (size_t)b * 2 * NBINS;
    const unsigned* hb = hf + NBINS;
    long long fgc = 0, bgc = 0;
    for (int i = 0; i < NBINS; ++i) { fgc += hf[i]; bgc += hb[i]; }
    long long keptfg = (fgc < NUM_FG_K) ? fgc : NUM_FG_K;
    long long k, total;
    const unsigned* h;
    if (cls == 0) { k = NUM_FG_K; h = hf; total = fgc; }
    else          { k = RPN_BATCH - keptfg; h = hb; total = bgc; }
    if (cls == 1 && b == 0) {
      long long keptbg = (bgc < k) ? bgc : k;
      uniw[0] = 1.0f / (float)(keptfg + keptbg);
    }
    if (total <= k) {
      thr[b * 2 + cls] = NEG_INF_F;     // keep all (reference NEG_INF case)
      s_done = 1;
    } else {
      long long cum = 0; int bkt = NBINS - 1;
      for (int i = NBINS - 1; i >= 0; --i) {
        if (cum + (long long)h[i] >= k) { bkt = i; break; }
        cum += h[i];
      }
      s_bucket = bkt;
      s_kprime = (int)(k - cum);        // rank within the bucket (1-indexed)
      s_done = 0;
      s_cnt = 0u;
    }
  }
  __syncthreads();
  if (s_done) return;

  int HW = H * W, NA = A * HW;
  int want = (cls == 0) ? 1 : 0;
  int bkt = s_bucket;
  for (int m = threadIdx.x; m < NA; m += blockDim.x) {
    if (labels[(size_t)b * NA + m] != want) continue;
    int a, y, x; decode_m(m, HW, W, hwsh, wsh, a, y, x);
    float s = rs[(size_t)b * NA + (size_t)(y * W + x) * A + a];
    int bin = (int)(s * (float)NBINS);
    if (bin > NBINS - 1) bin = NBINS - 1;
    if (bin < 0) bin = 0;
    if (bin == bkt) {
      unsigned idx = atomicAdd(&s_cnt, 1u);
      if (idx < CAND_CAP) cand[idx] = s;
    }
  }
  __syncthreads();
  if (threadIdx.x == 0) {
    int cnt = (int)((s_cnt < (unsigned)CAND_CAP) ? s_cnt : (unsigned)CAND_CAP);
    int kp = s_kprime; if (kp > cnt) kp = cnt;
    float cur = NEG_INF_F;
    for (int it = 0; it < kp; ++it) {        // exact k'-th largest in bucket
      float mx = -2.0f; int mi = 0;
      for (int j = 0; j < cnt; ++j)
        if (cand[j] > mx) { mx = cand[j]; mi = j; }
      cand[mi] = -3.0f;
      cur = mx;
    }
    thr[b * 2 + cls] = cur;
  }
}

// ---------------------------------------------------------------------------
// K5: final labels after subsampling, bbox targets, inside/outside weights,
// all written directly into the transposed output layouts (coalesced in x).
// grid = (ceil(NA/256), B)
// ---------------------------------------------------------------------------
__global__ void k5_out(const float* __restrict__ gt,
                       const float* __restrict__ anchors,
                       const float* __restrict__ im_info,
                       const int* __restrict__ argm,
                       const int* __restrict__ labels,
                       const float* __restrict__ rs,
                       const float* __restrict__ thr,
                       const float* __restrict__ uniw,
                       float* __restrict__ out,
                       int B, int G, int A, int H, int W, int hwsh, int wsh) {
  __shared__ float sgt[MAXG * 5];
  __shared__ float sanc[MAXA * 4];
  int b = blockIdx.y;
  stage_lds(sgt, gt + (size_t)b * G * 5, G * 5);
  stage_lds(sanc, anchors, A * 4);
  wait_stage();
  __syncthreads();

  int HW = H * W, NA = A * HW;
  int m = blockIdx.x * blockDim.x + threadIdx.x;
  if (m >= NA) return;
  int a, y, x; decode_m(m, HW, W, hwsh, wsh, a, y, x);
  size_t bm = (size_t)b * NA + m;

  int l = labels[bm];
  float s = rs[(size_t)b * NA + (size_t)(y * W + x) * A + a];
  float tf = thr[b * 2 + 0], tb = thr[b * 2 + 1];
  if (l == 1 && s < tf) l = -1;   // fg subsample
  if (l == 0 && s < tb) l = -1;   // bg subsample
  float uw = uniw[0];

  float ax1, ay1, ax2, ay2, aw, ah, aarea;
  anchor_box(sanc, a, x, y, ax1, ay1, ax2, ay2, aw, ah, aarea);
  float im_h = im_info[0], im_w = im_info[1];
  bool inside = (ax1 >= 0.0f) && (ay1 >= 0.0f) && (ax2 < im_w) && (ay2 < im_h);

  int g = argm[bm];
  float g0 = sgt[g * 5 + 0], g1 = sgt[g * 5 + 1];
  float g2 = sgt[g * 5 + 2], g3 = sgt[g * 5 + 3];
  float tgw = g2 - g0 + 1.0f, tgh = g3 - g1 + 1.0f;
  float ex_cx = ax1 + 0.5f * (aw - 1.0f), ex_cy = ay1 + 0.5f * (ah - 1.0f);
  float tgcx = g0 + 0.5f * (tgw - 1.0f), tgcy = g1 + 0.5f * (tgh - 1.0f);
  float t0 = (tgcx - ex_cx) / aw;
  float t1 = (tgcy - ex_cy) / ah;
  float t2 = logf(tgw / aw);
  float t3 = logf(tgh / ah);
  if (!inside) { t0 = 0.0f; t1 = 0.0f; t2 = 0.0f; t3 = 0.0f; }  // unmap fill=0

  size_t HWs = (size_t)HW;
  size_t totA = (size_t)B * A * HWs;
  float* lab_o = out;
  float* bt_o  = out + totA;
  float* biw_o = bt_o + 4 * totA;
  float* bow_o = biw_o + 4 * totA;

  lab_o[bm] = (float)l;                                  // [B,1,A*H,W]
  size_t base = (((size_t)b * 4 * A + 4 * a) * H + y) * W + x;   // [B,4A,H,W]
  bt_o[base + 0 * HWs] = t0;
  bt_o[base + 1 * HWs] = t1;
  bt_o[base + 2 * HWs] = t2;
  bt_o[base + 3 * HWs] = t3;
  float iw = (l == 1) ? 1.0f : 0.0f;
  biw_o[base + 0 * HWs] = iw; biw_o[base + 1 * HWs] = iw;
  biw_o[base + 2 * HWs] = iw; biw_o[base + 3 * HWs] = iw;
  float ow = (l >= 0) ? uw : 0.0f;
  bow_o[base + 0 * HWs] = ow; bow_o[base + 1 * HWs] = ow;
  bow_o[base + 2 * HWs] = ow; bow_o[base + 3 * HWs] = ow;
}

// ---------------------------------------------------------------------------
extern "C" void kernel_launch(void* const* d_in, const int* in_sizes, int n_in,
                              void* d_out, int out_size, void* d_ws, size_t ws_size,
                              hipStream_t stream) {
  const float* gt      = (const float*)d_in[1];   // [B,G,5]
  const float* im_info = (const float*)d_in[2];   // [B,3]
  const float* anchors = (const float*)d_in[3];   // [A,4]
  const float* rs      = (const float*)d_in[4];   // [B, H*W*A]
  // rpn_cls_score (d_in[0]) is shape-only in the reference; never read.

  int A = in_sizes[3] / 4;
  int B = in_sizes[2] / 3;
  int G = in_sizes[1] / (5 * B);
  int HW = (in_sizes[4] / B) / A;
  int H = 1; while ((long long)H * H < (long long)HW) ++H;   // square feature map
  int W = HW / H;
  int NA = A * H * W;

  // pow2 fast-path shifts for the device-side (a,y,x) decode
  int wsh = -1, hwsh = -1;
  if ((W & (W - 1)) == 0)  { wsh = 0;  while ((1 << wsh)  < W)  ++wsh;  }
  if ((HW & (HW - 1)) == 0){ hwsh = 0; while ((1 << hwsh) < HW) ++hwsh; }

  // scratch layout in d_ws
  float*    d_maxov = (float*)d_ws;                 // B*NA floats
  int*      d_arg   = (int*)(d_maxov + (size_t)B * NA);
  int*      d_lab   = (int*)(d_arg + (size_t)B * NA);
  unsigned* d_gtmax = (unsigned*)(d_lab + (size_t)B * NA);
  unsigned* d_hist  = d_gtmax + (size_t)B * G;      // B*2*NBINS
  float*    d_thr   = (float*)(d_hist + (size_t)B * 2 * NBINS);  // B*2
  float*    d_uni   = d_thr + (size_t)B * 2;        // 1

  int ngt = B * G, nhist = B * 2 * NBINS;
  int ninit = (ngt > nhist) ? ngt : nhist;
  k0_init<<<(ninit + 255) / 256, 256, 0, stream>>>(d_gtmax, d_hist, ngt, nhist);

  dim3 gridN((NA + 255) / 256, B);
  k1_maxarg<<<gridN, 256, 0, stream>>>(gt, anchors, im_info, d_maxov, d_arg,
                                       G, A, H, W, hwsh, wsh);
  k1b_gtmax<<<B * G * GPARTS, 256, 0, stream>>>(gt, anchors, im_info, d_gtmax,
                                                G, A, H, W, hwsh, wsh);
  k2_labels<<<gridN, 256, 0, stream>>>(gt, anchors, im_info, d_maxov, d_gtmax,
                                       rs, d_lab, d_hist, G, A, H, W, hwsh, wsh);
  k3_thresh<<<dim3(B, 2), 1024, 0, stream>>>(d_hist, rs, d_lab, d_thr, d_uni,
                                             A, H, W, hwsh, wsh);
  k5_out<<<gridN, 256, 0, stream>>>(gt, anchors, im_info, d_arg, d_lab, rs,
                                    d_thr, d_uni, (float*)d_out,
                                    B, G, A, H, W, hwsh, wsh);
}